// KitNET_5257039970983
// MI455X (gfx1250) — compile-verified
//
#include <hip/hip_runtime.h>

typedef float v2f __attribute__((ext_vector_type(2)));
typedef float v4f __attribute__((ext_vector_type(4)));
typedef float v8f __attribute__((ext_vector_type(8)));

#define B_TOTAL 500000
#define F_DIM   100
#define NT      10
#define C_DIM   10
#define H_DIM   7

#define ROWS_PER_BLOCK    80
#define THREADS_PER_BLOCK 160   // 5 waves (wave32)
#define NUM_WAVES         5

#define XCS_STRIDE 124          // 10 tails * 12 padded cols, +4 pad for banks
#define ZT_STRIDE  20           // ztile row stride (8 rows h, 16 cols m used)
#define TLS_STRIDE 12           // 10 tails + [1.0, 0] bias columns

__device__ __forceinline__ v8f wmma_f32_4(v2f a, v2f b, v8f c) {
    // D(16x16) = A(16x4) * B(4x16) + C, fp32
    return __builtin_amdgcn_wmma_f32_16x16x4_f32(
        false, a, false, b, (short)0, c, false, false);
}

__global__ __launch_bounds__(THREADS_PER_BLOCK)
void kitnet_kernel(const float* __restrict__ x,
                   const float* __restrict__ Wt,
                   const float* __restrict__ hbias_t,
                   const float* __restrict__ vbias_t,
                   const float* __restrict__ Wh,
                   const float* __restrict__ hbias_h,
                   const float* __restrict__ vbias_h,
                   const int*   __restrict__ clusters,
                   float* __restrict__ out_head,
                   float* __restrict__ out_tails)
{
    // xc gathered+padded: [row][t*12 + c], c=10 -> 1.0 (bias lane), c=11 -> 0
    __shared__ __align__(16) float xcs  [ROWS_PER_BLOCK * XCS_STRIDE];  // 9920 f
    // einsum1 B: [t][n=h(16)][k=c(12)] = Wt[t,h,c]; c==10 -> hbias_t[t,h]
    __shared__ __align__(16) float wtb1 [NT * 16 * 12];                 // 1920 f
    // einsum2 (transposed) A: [t][m=c(16)][k=h(12, use 8)] = Wt[t,h,c]; h==7 -> vbias_t[t,c]
    __shared__ __align__(16) float wta2 [NT * 16 * 12];                 // 1920 f
    // head einsum1 B: [n=h(16)][k=t(12)] = Wh[h,t]; t==10 -> hbias_h[h]
    __shared__ __align__(16) float whb1 [16 * 12];                      // 192 f
    // head einsum2 A: [m=t'(16)][k=h(12, use 8)] = Wh[h,t']; h==7 -> vbias_h[t']
    __shared__ __align__(16) float whb2t[16 * 12];                      // 192 f
    // per-wave z-transpose tile: [h(8)][m], row h=7 preset to 1.0 (bias lane)
    __shared__ __align__(16) float zbuf [NUM_WAVES * 8 * ZT_STRIDE];    // 800 f
    // tails tile: [row][TLS_STRIDE]; col 10 = 1.0, col 11 = 0
    __shared__ __align__(16) float tls  [ROWS_PER_BLOCK * TLS_STRIDE];  // 960 f

    const int tid = threadIdx.x;
    const int blockRow = blockIdx.x * ROWS_PER_BLOCK;

    // ---------------- staging ----------------
    for (int i = tid; i < NT * 16 * 12; i += THREADS_PER_BLOCK) {
        int t = i / 192, r = i % 192, h = r / 12, c = r % 12;
        float v = 0.f;
        if (h < H_DIM) {
            if (c < C_DIM)       v = Wt[t * 70 + h * 10 + c];
            else if (c == 10)    v = hbias_t[t * H_DIM + h];
        }
        wtb1[i] = v;
    }
    for (int i = tid; i < NT * 16 * 12; i += THREADS_PER_BLOCK) {
        int t = i / 192, r = i % 192, c = r / 12, h = r % 12;
        float v = 0.f;
        if (c < C_DIM) {
            if (h < H_DIM)       v = Wt[t * 70 + h * 10 + c];
            else if (h == 7)     v = vbias_t[t * C_DIM + c];
        }
        wta2[i] = v;
    }
    for (int i = tid; i < 16 * 12; i += THREADS_PER_BLOCK) {
        int h = i / 12, t = i % 12;
        float v = 0.f;
        if (h < H_DIM) {
            if (t < NT)          v = Wh[h * 10 + t];
            else if (t == 10)    v = hbias_h[h];
        }
        whb1[i] = v;
    }
    for (int i = tid; i < 16 * 12; i += THREADS_PER_BLOCK) {
        int tp = i / 12, h = i % 12;
        float v = 0.f;
        if (tp < NT) {
            if (h < H_DIM)       v = Wh[h * 10 + tp];
            else if (h == 7)     v = vbias_h[tp];
        }
        whb2t[i] = v;
    }
    // z-tile bias row (h=7) = 1.0, once per wave-tile
    for (int i = tid; i < NUM_WAVES * 16; i += THREADS_PER_BLOCK)
        zbuf[(i / 16) * (8 * ZT_STRIDE) + 7 * ZT_STRIDE + (i % 16)] = 1.0f;
    // tails bias columns
    for (int i = tid; i < ROWS_PER_BLOCK; i += THREADS_PER_BLOCK) {
        tls[i * TLS_STRIDE + 10] = 1.0f;
        tls[i * TLS_STRIDE + 11] = 0.0f;
    }
    // gather xc (cluster indices from global: 400B, cache-resident)
    for (int i = tid; i < ROWS_PER_BLOCK * NT * 12; i += THREADS_PER_BLOCK) {
        int row = i / (NT * 12), j = i % (NT * 12), t = j / 12, c = j % 12;
        float v;
        if (c < C_DIM)      v = x[(size_t)(blockRow + row) * F_DIM + clusters[t * C_DIM + c]];
        else if (c == 10)   v = 1.0f;
        else                v = 0.0f;
        xcs[row * XCS_STRIDE + j] = v;
    }
    __syncthreads();

    const int lane   = tid & 31;
    const int wave   = tid >> 5;
    const int half   = lane >> 4;
    const int lane16 = lane & 15;
    const int wrb    = wave * 16;
    float* ztile = &zbuf[wave * 8 * ZT_STRIDE];

    const float* myxc = &xcs[(wrb + lane16) * XCS_STRIDE];

    // ================= tail autoencoders =================
    for (int t = 0; t < NT; ++t) {
        // ---- z(16x8,+bias) = xc(16x12) * Wt1 : K=c padded 12 (col10=1 -> +hbias) ----
        v8f acc = {0.f,0.f,0.f,0.f,0.f,0.f,0.f,0.f};
        #pragma unroll
        for (int k = 0; k < 3; ++k) {
            const int c0 = 4 * k + 2 * half;
            v2f a = *(const v2f*)&myxc [t * 12 + c0];
            v2f b = *(const v2f*)&wtb1[t * 192 + lane16 * 12 + c0];
            acc = wmma_f32_4(a, b, acc);
        }
        // stash z transposed: ztile[h][m]; row 7 stays 1.0 (bias lane)
        if (lane16 < 7) {
            float* zp = &ztile[lane16 * ZT_STRIDE + 8 * half];
            *(v4f*)(zp)     = (v4f){acc[0], acc[1], acc[2], acc[3]};
            *(v4f*)(zp + 4) = (v4f){acc[4], acc[5], acc[6], acc[7]};
        }

        // ---- out^T(16x16) = WtA2(16x8) * z^T(8x16) : K=h padded 8 (row7 -> +vbias) ----
        v8f acc2 = {0.f,0.f,0.f,0.f,0.f,0.f,0.f,0.f};
        #pragma unroll
        for (int k = 0; k < 2; ++k) {
            const int c0 = 4 * k + 2 * half;
            v2f a = *(const v2f*)&wta2[t * 192 + lane16 * 12 + c0];
            v2f b;
            b.x = ztile[(c0    ) * ZT_STRIDE + lane16];
            b.y = ztile[(c0 + 1) * ZT_STRIDE + lane16];
            acc2 = wmma_f32_4(a, b, acc2);
        }

        // ---- rmse over c: in-lane sum + one cross-half shuffle ----
        const float* xcp = &myxc[t * 12 + 8 * half];
        v4f x0 = *(const v4f*)(xcp);
        v4f x1 = *(const v4f*)(xcp + 4);
        float d0 = acc2[0] - x0[0], d1 = acc2[1] - x0[1];
        float d2 = acc2[2] - x0[2], d3 = acc2[3] - x0[3];
        float d4 = acc2[4] - x1[0], d5 = acc2[5] - x1[1];
        float d6 = acc2[6] - x1[2], d7 = acc2[7] - x1[3];
        float s_low = d0*d0 + d1*d1;                         // c = 8,9 when half==1
        float s_all = s_low + d2*d2 + d3*d3 + d4*d4 + d5*d5 + d6*d6 + d7*d7;
        float sq = (half == 0) ? s_all : s_low;
        sq += __shfl_xor(sq, 16, 32);
        const float tv = 0.5f * logf(0.1f * sq);             // log(sqrt(mean))
        if (half == 0)
            tls[(wrb + lane16) * TLS_STRIDE + t] = tv;
    }

    // ================= head autoencoder on tails =================
    {
        const float* trow = &tls[(wrb + lane16) * TLS_STRIDE];
        v8f acc = {0.f,0.f,0.f,0.f,0.f,0.f,0.f,0.f};
        #pragma unroll
        for (int k = 0; k < 3; ++k) {
            const int c0 = 4 * k + 2 * half;
            v2f a = *(const v2f*)&trow[c0];                  // col10=1 -> +hbias_h
            v2f b = *(const v2f*)&whb1[lane16 * 12 + c0];
            acc = wmma_f32_4(a, b, acc);
        }
        if (lane16 < 7) {
            float* zp = &ztile[lane16 * ZT_STRIDE + 8 * half];
            *(v4f*)(zp)     = (v4f){acc[0], acc[1], acc[2], acc[3]};
            *(v4f*)(zp + 4) = (v4f){acc[4], acc[5], acc[6], acc[7]};
        }
        v8f acc2 = {0.f,0.f,0.f,0.f,0.f,0.f,0.f,0.f};
        #pragma unroll
        for (int k = 0; k < 2; ++k) {
            const int c0 = 4 * k + 2 * half;
            v2f a = *(const v2f*)&whb2t[lane16 * 12 + c0];   // row7 -> +vbias_h
            v2f b;
            b.x = ztile[(c0    ) * ZT_STRIDE + lane16];
            b.y = ztile[(c0 + 1) * ZT_STRIDE + lane16];
            acc2 = wmma_f32_4(a, b, acc2);
        }
        // D holds head_out^T: rows t' = r + 8*half, col m = lane16
        const size_t ob = (size_t)(blockRow + wrb + lane16) * NT;
        if (half == 0) {
            *(v2f*)&out_head[ob + 0] = (v2f){acc2[0], acc2[1]};
            *(v2f*)&out_head[ob + 2] = (v2f){acc2[2], acc2[3]};
            *(v2f*)&out_head[ob + 4] = (v2f){acc2[4], acc2[5]};
            *(v2f*)&out_head[ob + 6] = (v2f){acc2[6], acc2[7]};
        } else {
            *(v2f*)&out_head[ob + 8] = (v2f){acc2[0], acc2[1]};  // t' = 8,9
        }
    }

    // ================= coalesced tails write-out =================
    __syncthreads();
    for (int i = tid; i < ROWS_PER_BLOCK * NT; i += THREADS_PER_BLOCK) {
        int row = i / NT, t = i % NT;
        out_tails[(size_t)blockRow * NT + i] = tls[row * TLS_STRIDE + t];
    }
}

extern "C" void kernel_launch(void* const* d_in, const int* in_sizes, int n_in,
                              void* d_out, int out_size, void* d_ws, size_t ws_size,
                              hipStream_t stream) {
    const float* x   = (const float*)d_in[0];
    const float* Wt  = (const float*)d_in[1];
    const float* hbt = (const float*)d_in[2];
    const float* vbt = (const float*)d_in[3];
    const float* Wh  = (const float*)d_in[4];
    const float* hbh = (const float*)d_in[5];
    const float* vbh = (const float*)d_in[6];
    const int*   clu = (const int*)d_in[7];
    float* out = (float*)d_out;

    dim3 grid(B_TOTAL / ROWS_PER_BLOCK);   // 6250 blocks, exact
    kitnet_kernel<<<grid, THREADS_PER_BLOCK, 0, stream>>>(
        x, Wt, hbt, vbt, Wh, hbh, vbh, clu,
        out, out + (size_t)B_TOTAL * NT);
}